// GraphSAGE_52218212384880
// MI455X (gfx1250) — compile-verified
//
#include <hip/hip_runtime.h>
#include <math.h>

#define NODES 100000
#define EDGES 800000
#define FDIM  128
#define OUTD  40
#define BN_EPS 1e-5f
#define LDSROW 132   // 16 tile rows, padded stride -> conflict-free b64 reads

typedef float v2f __attribute__((ext_vector_type(2)));
typedef float v8f __attribute__((ext_vector_type(8)));

// ---------------------------------------------------------------- utilities
__global__ void k_fill0(float* __restrict__ p, int n) {
    int i = blockIdx.x * blockDim.x + threadIdx.x;
    int stride = gridDim.x * blockDim.x;
    for (; i < n; i += stride) p[i] = 0.0f;
}

// in-degree count (float so we can reuse the same atomic path)
__global__ void k_count(const int* __restrict__ ei, float* __restrict__ cnt) {
    int e = blockIdx.x * blockDim.x + threadIdx.x;
    if (e >= EDGES) return;
    int dst = ei[EDGES + e];
    atomicAdd(&cnt[dst], 1.0f);
}

__global__ void k_invert(float* __restrict__ c) {
    int i = blockIdx.x * blockDim.x + threadIdx.x;
    if (i >= NODES) return;
    c[i] = 1.0f / fmaxf(c[i], 1.0f);
}

// ------------------------------------------------------- edge scatter (mean)
// 32 threads per edge: thread t of the edge-group handles channels 4t..4t+3.
// A full wave covers one edge's 128 channels = 512 contiguous bytes.
__global__ void k_aggregate(const float* __restrict__ feat, const int* __restrict__ ei,
                            float* __restrict__ agg) {
    int t = blockIdx.x * blockDim.x + threadIdx.x;
    int e = t >> 5;
    if (e >= EDGES) return;
    int c = (t & 31) * 4;
    int src = ei[e];
    int dst = ei[EDGES + e];
    const float4 v = *(const float4*)(feat + (size_t)src * FDIM + c);
    float* d = agg + (size_t)dst * FDIM + c;
    atomicAdd(d + 0, v.x);
    atomicAdd(d + 1, v.y);
    atomicAdd(d + 2, v.z);
    atomicAdd(d + 3, v.w);
}

// --------------------------------------------------------------- WMMA GEMM
// out[r, n] = sum_k (agg[r,k]*inv[r]) * Wl[k,n] + sum_k x[r,k] * Wr[k,n] + bl[n]
// Block = 4 waves, one 16x16 tile per wave, all sharing one column tile.
// Weight tiles are staged (transposed + zero-padded) into LDS once per block,
// so the fully-unrolled K loop is branch-free: per step one global b64 (A),
// one ds b64 (B), one v_wmma_f32_16x16x4_f32.
__global__ void __launch_bounds__(128)
k_sage_gemm(const float* __restrict__ agg, const float* __restrict__ inv_cnt,
            const float* __restrict__ x,
            const float* __restrict__ Wl, const float* __restrict__ bl,
            const float* __restrict__ Wr,
            float* __restrict__ out, float* __restrict__ stats,
            int outF) {
    __shared__ float Blt[16 * LDSROW];   // Blt[n][k] = Wl[k][col0+n] (0 padded)
    __shared__ float Brt[16 * LDSROW];   // Brt[n][k] = Wr[k][col0+n] (0 padded)

    const int tid  = threadIdx.x;        // 128 threads: thread t owns k-row t
    const int col0 = blockIdx.x * 16;
    {
        const size_t rowOff = (size_t)tid * outF;
#pragma unroll 4
        for (int n = 0; n < 16; ++n) {
            int nc = col0 + n;
            float vl = 0.0f, vr = 0.0f;
            if (nc < outF) { vl = Wl[rowOff + nc]; vr = Wr[rowOff + nc]; }
            Blt[n * LDSROW + tid] = vl;
            Brt[n * LDSROW + tid] = vr;
        }
    }
    __syncthreads();

    const int lane    = tid & 31;
    const int wave    = tid >> 5;
    const int rowTile = blockIdx.y * 4 + wave;
    const int row0    = rowTile * 16;
    if (row0 < NODES) {                      // wave-uniform: EXEC all-ones inside
        const int half = lane >> 4;          // 0: K={0,1}   1: K={2,3}
        const int ml   = lane & 15;
        const int arow = row0 + ml;          // A-row carried by this lane
        const int ncol = col0 + ml;          // B/D column carried by this lane
        const float inv = inv_cnt[arow];

        const float* Ag  = agg + (size_t)arow * FDIM + 2 * half;
        const float* Xg  = x   + (size_t)arow * FDIM + 2 * half;
        const float* Bls = &Blt[ml * LDSROW + 2 * half];
        const float* Brs = &Brt[ml * LDSROW + 2 * half];

        v8f acc = {};
        // ---- pass 1: (agg * inv_cnt) @ Wl
#pragma unroll
        for (int kb = 0; kb < FDIM; kb += 4) {
            float2 av = *(const float2*)(Ag + kb);
            float2 bv = *(const float2*)(Bls + kb);
            v2f a; a.x = av.x * inv; a.y = av.y * inv;
            v2f b; b.x = bv.x;       b.y = bv.y;
            acc = __builtin_amdgcn_wmma_f32_16x16x4_f32(false, a, false, b,
                                                        (short)0, acc, false, false);
        }
        // ---- pass 2: x @ Wr (same accumulator)
#pragma unroll
        for (int kb = 0; kb < FDIM; kb += 4) {
            float2 av = *(const float2*)(Xg + kb);
            float2 bv = *(const float2*)(Brs + kb);
            v2f a; a.x = av.x; a.y = av.y;
            v2f b; b.x = bv.x; b.y = bv.y;
            acc = __builtin_amdgcn_wmma_f32_16x16x4_f32(false, a, false, b,
                                                        (short)0, acc, false, false);
        }

        // ---- epilogue: bias, store, fused BN column statistics
        const bool colOk = (ncol < outF);
        const float bias = colOk ? bl[ncol] : 0.0f;
        float s = 0.0f, sq = 0.0f;
#pragma unroll
        for (int v = 0; v < 8; ++v) {
            float val = acc[v] + bias;       // C layout: row = row0 + v + 8*half
            s += val; sq += val * val;
            if (colOk) out[(size_t)(row0 + v + 8 * half) * outF + ncol] = val;
        }
        if (stats != nullptr) {
            // lanes l and l+16 hold the same column (rows 0-7 vs 8-15)
            s  += __shfl_xor(s, 16, 32);
            sq += __shfl_xor(sq, 16, 32);
            if (half == 0 && colOk) {
                atomicAdd(&stats[ncol], s);
                atomicAdd(&stats[FDIM + ncol], sq);
            }
        }
    }
}

// ------------------------------------------------------------- BatchNorm
__global__ void k_bn_prep(const float* __restrict__ stats, const float* __restrict__ g,
                          const float* __restrict__ b, float* __restrict__ ss) {
    int c = threadIdx.x;                     // 128 threads
    const float invN = 1.0f / (float)NODES;
    float mean = stats[c] * invN;
    float var  = stats[FDIM + c] * invN - mean * mean;   // biased variance
    float sc   = g[c] * rsqrtf(var + BN_EPS);
    ss[c]        = sc;
    ss[FDIM + c] = b[c] - mean * sc;
}

__global__ void k_bn_relu(float* __restrict__ h, const float* __restrict__ ss, int total) {
    int i = blockIdx.x * blockDim.x + threadIdx.x;
    if (i >= total) return;
    int c = i & (FDIM - 1);
    float v = h[i] * ss[c] + ss[FDIM + c];
    h[i] = v > 0.0f ? v : 0.0f;
}

// ------------------------------------------------------------ log-softmax
__global__ void k_logsoftmax(const float* __restrict__ in, float* __restrict__ out) {
    int r = blockIdx.x * blockDim.x + threadIdx.x;
    if (r >= NODES) return;
    const float* p = in + (size_t)r * OUTD;
    float mx = -3.0e38f;
    for (int i = 0; i < OUTD; ++i) mx = fmaxf(mx, p[i]);
    float sum = 0.0f;
    for (int i = 0; i < OUTD; ++i) sum += __expf(p[i] - mx);
    float lse = mx + __logf(sum);
    float* q = out + (size_t)r * OUTD;
    for (int i = 0; i < OUTD; ++i) q[i] = p[i] - lse;
}

// ---------------------------------------------------------------- launcher
extern "C" void kernel_launch(void* const* d_in, const int* in_sizes, int n_in,
                              void* d_out, int out_size, void* d_ws, size_t ws_size,
                              hipStream_t stream) {
    (void)in_sizes; (void)n_in; (void)out_size; (void)ws_size;
    const float* x  = (const float*)d_in[0];
    const int*   ei = (const int*)d_in[1];
    const float* Wl[4] = {(const float*)d_in[2], (const float*)d_in[5],
                          (const float*)d_in[8], (const float*)d_in[11]};
    const float* bl[4] = {(const float*)d_in[3], (const float*)d_in[6],
                          (const float*)d_in[9], (const float*)d_in[12]};
    const float* Wr[4] = {(const float*)d_in[4], (const float*)d_in[7],
                          (const float*)d_in[10], (const float*)d_in[13]};
    const float* g[3]  = {(const float*)d_in[14], (const float*)d_in[16], (const float*)d_in[18]};
    const float* b[3]  = {(const float*)d_in[15], (const float*)d_in[17], (const float*)d_in[19]};

    // workspace layout (floats): inv_cnt | agg | bufA | bufB | stats(256) | ss(256)
    float* ws = (float*)d_ws;
    const size_t N_AL = 100032;               // NODES rounded to 64 floats
    const size_t NF   = (size_t)NODES * FDIM; // 12.8M floats
    float* invcnt = ws;
    float* agg    = invcnt + N_AL;
    float* bufA   = agg + NF;
    float* bufB   = bufA + NF;
    float* stats  = bufB + NF;
    float* ss     = stats + 2 * FDIM;

    // --- in-degree -> 1/max(deg,1)
    k_fill0 <<<256, 256, 0, stream>>>(invcnt, NODES);
    k_count <<<(EDGES + 255) / 256, 256, 0, stream>>>(ei, invcnt);
    k_invert<<<(NODES + 255) / 256, 256, 0, stream>>>(invcnt);

    const int rowBlocks = (NODES / 16 + 3) / 4;   // 1563 (4 waves / block)

    auto layer = [&](const float* xin, int li, float* hout, bool bn, int outF) {
        k_fill0<<<2048, 256, 0, stream>>>(agg, (int)NF);
        k_aggregate<<<(EDGES * 32 + 255) / 256, 256, 0, stream>>>(xin, ei, agg);
        float* st = nullptr;
        if (bn) { k_fill0<<<1, 256, 0, stream>>>(stats, 2 * FDIM); st = stats; }
        dim3 grid((outF + 15) / 16, rowBlocks);
        k_sage_gemm<<<grid, 128, 0, stream>>>(agg, invcnt, xin, Wl[li], bl[li], Wr[li],
                                              hout, st, outF);
        if (bn) {
            k_bn_prep<<<1, FDIM, 0, stream>>>(stats, g[li], b[li], ss);
            k_bn_relu<<<(int)((NF + 255) / 256), 256, 0, stream>>>(hout, ss, (int)NF);
        }
    };

    layer(x,    0, bufA, true,  FDIM);   // conv1 + BN + ReLU
    layer(bufA, 1, bufB, true,  FDIM);   // conv2 + BN + ReLU
    layer(bufB, 2, bufA, true,  FDIM);   // conv3 + BN + ReLU
    layer(bufA, 3, bufB, false, OUTD);   // conv4 (pre-softmax, stride 40)

    k_logsoftmax<<<(NODES + 255) / 256, 256, 0, stream>>>(bufB, (float*)d_out);
}